// RTDETR_postprocess_without_score_scaling_76656576299740
// MI455X (gfx1250) — compile-verified
//
#include <hip/hip_runtime.h>
#include <cstdint>
#include <cstddef>

// RT-DETR postprocess (threshold -> stable sort -> greedy NMS -> masked emit),
// N=4096. MI455X/gfx1250: 2.2MB working set lives entirely in L2 (192MB) ->
// latency-bound, not bandwidth/FLOP bound. CDNA5-specific paths:
//   - tensor_load_to_lds (TDM, TENSORcnt) DMAs the column-box tile into LDS
//   - global_load_async_to_lds_b128 (ASYNCcnt) stages the row-box tile,
//     overlapping with the TDM transfer
//   - v_readlane_b32 broadcast on the serial greedy-NMS chain (wave32, no
//     barriers, no LDS round-trip) + global_prefetch_b8 to hide L2 latency

#define N 4096
#define IMG 640.0f

typedef unsigned int v4u __attribute__((ext_vector_type(4)));
typedef int          v8i __attribute__((ext_vector_type(8)));
typedef int          v4i __attribute__((ext_vector_type(4)));

// ---------------------------------------------------------------------------
// Kernel 1: one block, 1024 threads.
// 64-bit keys [ordered conf bits | (N-1-i)] -> descending bitonic sort in LDS
// == stable argsort(-key); invalid rows (conf < 0.5) get key-high 0 (-inf).
// Fused gather emits sorted xyxy float4 boxes, sorted conf, and valid count V.
// ---------------------------------------------------------------------------
__global__ __launch_bounds__(1024) void prep_sort_kernel(
    const float* __restrict__ raw, float4* __restrict__ boxesS,
    float* __restrict__ confS, int* __restrict__ vcnt) {
  __shared__ unsigned long long skeys[N];
  __shared__ int vcounter;
  const int tid = threadIdx.x;
  if (tid == 0) vcounter = 0;

  #pragma unroll
  for (int m = 0; m < 4; ++m) {
    int i = tid + m * 1024;
    float conf = raw[i * 5 + 4];
    bool valid = (conf >= 0.5f);           // conf>0 implied by conf>=0.5
    unsigned u = valid ? (__float_as_uint(conf) | 0x80000000u) : 0u;
    skeys[i] = ((unsigned long long)u << 32) | (unsigned)(N - 1 - i);
  }
  __syncthreads();

  // Bitonic sort, descending; pairs {i, i^j} are disjoint per pass, only the
  // lower index acts -> race-free.
  for (int k = 2; k <= N; k <<= 1) {
    for (int j = k >> 1; j > 0; j >>= 1) {
      #pragma unroll
      for (int m = 0; m < 4; ++m) {
        int i = tid + m * 1024;
        int ixj = i ^ j;
        if (ixj > i) {
          unsigned long long a = skeys[i], b = skeys[ixj];
          bool up = ((i & k) == 0);
          if (up ? (a < b) : (a > b)) { skeys[i] = b; skeys[ixj] = a; }
        }
      }
      __syncthreads();
    }
  }

  int myValid = 0;
  #pragma unroll
  for (int m = 0; m < 4; ++m) {
    int i = tid + m * 1024;
    unsigned long long key = skeys[i];
    int orig = (N - 1) - (int)(key & 0xFFFFFFFFu);
    const float* p = raw + (size_t)orig * 5;
    float cx = p[0] * IMG, cy = p[1] * IMG;
    float w  = p[2] * IMG, h  = p[3] * IMG;
    boxesS[i] = make_float4(cx - 0.5f * w, cy - 0.5f * h,
                            cx + 0.5f * w, cy + 0.5f * h);
    confS[i] = p[4];
    myValid += ((unsigned)(key >> 32) != 0u) ? 1 : 0;
  }
  if (myValid) atomicAdd(&vcounter, myValid);
  __syncthreads();
  if (tid == 0) *vcnt = vcounter;
}

// ---------------------------------------------------------------------------
// Kernel 2: suppression bitmask, grid (32 col-tiles x 16 row-tiles), 256 thr.
// Per block: 256 rows x 128 cols. Column tile (128 x float4 = 2KB) arrives by
// ONE tensor_load_to_lds DMA; row tile (256 x float4 = 4KB) is staged by
// per-thread async LDS loads that overlap the TDM. Division-free IoU test:
// inter/max(union,1e-9) > 0.5  <=>  inter > 0.5*max(union,1e-9).
// Output: uint4 of mask words per row (bit only when j > r).
// ---------------------------------------------------------------------------
__global__ __launch_bounds__(256) void nms_mask_kernel(
    const float4* __restrict__ boxesS, const int* __restrict__ vcnt,
    unsigned* __restrict__ mask) {
  __shared__ float4 colBox[128];
  __shared__ float4 rowBox[256];
  const int V = *vcnt;
  const int rowBase = blockIdx.y * 256;
  if (rowBase >= V) return;                 // rows >= V never consulted
  const int colBase = blockIdx.x * 128;
  const int tid = threadIdx.x;

  // Stage row tile: each thread async-copies its own box (16B) to LDS.
  {
    unsigned lds_off = (unsigned)(uintptr_t)(&rowBox[tid]);
    unsigned long long ga =
        (unsigned long long)(uintptr_t)(boxesS + rowBase + tid);
    asm volatile("global_load_async_to_lds_b128 %0, %1, off"
                 :: "v"(lds_off), "v"(ga) : "memory");
  }

  // Stage column tile: one TDM descriptor, issued once by wave 0.
  // D# per CDNA5 ISA 8.3/8.4: 1D tile of 256 x 8B elements (= 128 float4).
#if __has_builtin(__builtin_amdgcn_tensor_load_to_lds)
  if (tid == 0) {
    unsigned long long ga =
        (unsigned long long)(uintptr_t)(boxesS + colBase);
    unsigned lds_off = (unsigned)(uintptr_t)(&colBox[0]);
    v4u g0;
    g0.x = 1u;                                   // count=1, no gather/restore
    g0.y = lds_off;                              // lds_addr [63:32]
    g0.z = (unsigned)(ga & 0xFFFFFFFFu);         // global_addr[31:0]
    g0.w = (unsigned)((ga >> 32) & 0x01FFFFFFu)  // global_addr[56:32]
           | (2u << 30);                         // type=2 (image)
    v8i g1;
    g1[0] = (3 << 16);          // wg_mask=0, data_size=3 (8B), no pad/iter
    g1[1] = (int)(256u << 16);  // tensor_dim0[15:0]=256 (bits 79:48)
    g1[2] = (int)(1u << 16);    // tensor_dim0 hi=0, tensor_dim1=1 (111:80)
    g1[3] = (int)(256u << 16);  // tensor_dim1 hi=0, tile_dim0=256 (127:112)
    g1[4] = 1;                  // tile_dim1=1, tile_dim2=0
    g1[5] = 256;                // tensor_dim0_stride lo (207:160)
    g1[6] = 0;                  // stride hi=0, tensor_dim1_stride lo=0
    g1[7] = 0;
    v4i g2;                     // degenerate upper dims
    g2[0] = 1;                  // tensor_dim2=1
    g2[1] = 1;                  // tensor_dim3=1
    g2[2] = 0;                  // tensor_dim2_stride lo
    g2[3] = 0;                  // stride hi, tile_dim3=0
    v4i g3;
    g3[0] = 0; g3[1] = (int)(1u << 16);          // tensor_dim4=1
    g3[2] = 0; g3[3] = 0;                        // tile_dim4=0
    // clang-23 / therock-10.0 lane: 6-arg form
    // (v4u, v8i, v4i, v4i, v8i, i32 cpol). Extra v8i carries no info for a
    // degenerate 1-D tile (gather/iterate disabled) -> zeros.
    v8i g4 = {0, 0, 0, 0, 0, 0, 0, 0};
    __builtin_amdgcn_tensor_load_to_lds(g0, g1, g2, g3, g4, 0);
  }
#if __has_builtin(__builtin_amdgcn_s_wait_tensorcnt)
  __builtin_amdgcn_s_wait_tensorcnt(0);
#else
  asm volatile("s_wait_tensorcnt 0" ::: "memory");
#endif
#else
  // Fallback: async-copy the column tile too.
  if (tid < 128) {
    unsigned lds_off = (unsigned)(uintptr_t)(&colBox[tid]);
    unsigned long long ga =
        (unsigned long long)(uintptr_t)(boxesS + colBase + tid);
    asm volatile("global_load_async_to_lds_b128 %0, %1, off"
                 :: "v"(lds_off), "v"(ga) : "memory");
  }
#endif
  asm volatile("s_wait_asynccnt 0" ::: "memory");
  __syncthreads();

  const int r = rowBase + tid;
  float4 rb = rowBox[tid];
  float areaR = (rb.z - rb.x) * (rb.w - rb.y);
  unsigned bw[4];
  for (int w = 0; w < 4; ++w) {
    unsigned bits = 0u;
    #pragma unroll
    for (int b = 0; b < 32; ++b) {
      int c = w * 32 + b;
      float4 cb = colBox[c];
      float areaC = (cb.z - cb.x) * (cb.w - cb.y);
      float ix1 = fmaxf(rb.x, cb.x), iy1 = fmaxf(rb.y, cb.y);
      float ix2 = fminf(rb.z, cb.z), iy2 = fminf(rb.w, cb.w);
      float iw = fmaxf(ix2 - ix1, 0.0f), ih = fmaxf(iy2 - iy1, 0.0f);
      float inter = iw * ih;
      float uni = fmaxf(areaR + areaC - inter, 1e-9f);
      int j = colBase + c;
      if ((inter > 0.5f * uni) && (j > r)) bits |= (1u << b);
    }
    bw[w] = bits;
  }
  // 16B-aligned: r*512B + blockIdx.x*16B
  *(uint4*)(mask + (size_t)r * (N / 32) + blockIdx.x * 4) =
      make_uint4(bw[0], bw[1], bw[2], bw[3]);
}

// ---------------------------------------------------------------------------
// Kernel 3: greedy reduce on ONE wave32 (zero barriers on the serial chain),
// then 8 waves emit the 4096x5 output with coalesced b32 stores.
// Lane l owns mask words 4l..4l+3 (one uint4 per surviving row). The
// "row i still kept" word is broadcast via v_readlane_b32 (owner is uniform).
// ---------------------------------------------------------------------------
__global__ __launch_bounds__(256) void nms_reduce_write_kernel(
    const float4* __restrict__ boxesS, const float* __restrict__ confS,
    const int* __restrict__ vcnt, const unsigned* __restrict__ mask,
    float* __restrict__ out) {
  __shared__ unsigned keepW[N / 32];
  const int tid = threadIdx.x;
  const int V = *vcnt;

  if (tid < 32) {
    unsigned r0 = 0, r1 = 0, r2 = 0, r3 = 0;
    for (int i = 0; i < V; ++i) {
      if (i + 4 < V)  // global_prefetch_b8 toward this WGP
        __builtin_prefetch(mask + (size_t)(i + 4) * (N / 32) + tid * 4, 0, 1);
      int w = i >> 5;                        // uniform
      int owner = w >> 2, slot = w & 3;      // uniform
      unsigned sel = (slot == 0) ? r0 : (slot == 1) ? r1
                   : (slot == 2) ? r2 : r3;
      unsigned word = (unsigned)__builtin_amdgcn_readlane((int)sel, owner);
      if (((word >> (i & 31)) & 1u) == 0u) {  // row i survives -> suppress
        const uint4* mrow = (const uint4*)(mask + (size_t)i * (N / 32));
        uint4 m = mrow[tid];
        r0 |= m.x; r1 |= m.y; r2 |= m.z; r3 |= m.w;
      }
    }
    unsigned rv[4] = {r0, r1, r2, r3};
    #pragma unroll
    for (int s = 0; s < 4; ++s) {
      int w = tid * 4 + s;
      int lo = w * 32;
      unsigned vm = (V >= lo + 32) ? 0xFFFFFFFFu
                  : (V <= lo)      ? 0u
                                   : ((1u << (V - lo)) - 1u);
      keepW[w] = (~rv[s]) & vm;   // keep = (prefix < V) && !suppressed
    }
  }
  __syncthreads();

  const float* bs = (const float*)boxesS;
  for (int e = tid; e < N * 5; e += 256) {
    int j = e / 5;
    int f = e - j * 5;
    bool keep = (keepW[j >> 5] >> (j & 31)) & 1u;
    float v = 0.0f;
    if (keep) v = (f < 4) ? bs[j * 4 + f] : confS[j];
    out[e] = v;
  }
}

// ---------------------------------------------------------------------------
// Host launcher. Workspace (~2.2MB):
//   [0, 64KB)          boxesS : float4[4096] sorted xyxy
//   [64KB, 80KB)       confS  : float[4096] sorted conf
//   [80KB, 80KB+4)     vcnt   : int, # valid rows
//   [128KB, 128KB+2MB) mask   : uint32[4096][128] suppression bits (j>i)
// ---------------------------------------------------------------------------
extern "C" void kernel_launch(void* const* d_in, const int* in_sizes, int n_in,
                              void* d_out, int out_size, void* d_ws,
                              size_t ws_size, hipStream_t stream) {
  (void)in_sizes; (void)n_in; (void)out_size; (void)ws_size;
  const float* raw = (const float*)d_in[0];
  float* out = (float*)d_out;
  char* ws = (char*)d_ws;
  float4*   boxesS = (float4*)(ws);
  float*    confS  = (float*)(ws + 65536);
  int*      vcnt   = (int*)(ws + 81920);
  unsigned* mask   = (unsigned*)(ws + 131072);

  prep_sort_kernel<<<1, 1024, 0, stream>>>(raw, boxesS, confS, vcnt);
  nms_mask_kernel<<<dim3(N / 128, N / 256), 256, 0, stream>>>(boxesS, vcnt,
                                                              mask);
  nms_reduce_write_kernel<<<1, 256, 0, stream>>>(boxesS, confS, vcnt, mask,
                                                 out);
}